// DempsterSchaferCombine_79113297592735
// MI455X (gfx1250) — compile-verified
//
#include <hip/hip_runtime.h>
#include <stdint.h>

// Dempster-Shafer combination of two Dirichlet opinions.
// Algebraic reduction (exact): the combined strength/denominator cancels and
//   alpha_a = (alpha1-1)*(alpha2-1)/C + alpha1 + alpha2 - 1   (C = 21)
// => purely elementwise, HBM-bandwidth bound (12 B/elem, ~528 MB total,
//    floor ~22.7us at 23.3 TB/s).
//
// Data path (gfx1250): double-buffered global_load_async_to_lds_b128 staging
// (ASYNCcnt), ds_load_b128 consume, nontemporal global_store_b128 output.

typedef float v4f __attribute__((ext_vector_type(4)));

#define BLOCK 256  // 8 wave32 waves per workgroup

__device__ __forceinline__ float ds_comb(float a, float b) {
    const float t1 = a - 1.0f;
    const float t2 = b - 1.0f;
    // (a-1)(b-1)/21 + (a-1) + (b-1) + 1
    return fmaf(t1 * t2, (1.0f / 21.0f), t1 + t2 + 1.0f);
}

__global__ __launch_bounds__(BLOCK) void ds_main(
    const float* __restrict__ a1, const float* __restrict__ a2,
    float* __restrict__ out, unsigned numTiles)
{
    __shared__ v4f sA[2][BLOCK];   // 2 x 4KB
    __shared__ v4f sB[2][BLOCK];   // 2 x 4KB  (16KB total)

    const unsigned tid = threadIdx.x;
    unsigned t = blockIdx.x;
    if (t >= numTiles) return;     // never taken when grid <= numTiles

    const uint64_t baseA = (uint64_t)(uintptr_t)a1;
    const uint64_t baseB = (uint64_t)(uintptr_t)a2;

    const unsigned step = gridDim.x * (BLOCK * (unsigned)sizeof(v4f));
    unsigned off = (t * BLOCK + tid) * (unsigned)sizeof(v4f);   // byte offset

    // LDS byte addresses (low 32 bits of the flat pointer == LDS offset)
    const unsigned lA0 = (unsigned)(uintptr_t)&sA[0][tid];
    const unsigned lA1 = (unsigned)(uintptr_t)&sA[1][tid];
    const unsigned lB0 = (unsigned)(uintptr_t)&sB[0][tid];
    const unsigned lB1 = (unsigned)(uintptr_t)&sB[1][tid];

    // Prologue: stage tile t into buffer 0 (async DMA, no VGPR data path)
    asm volatile("global_load_async_to_lds_b128 %0, %1, %2"
                 :: "v"(lA0), "v"(off), "s"(baseA) : "memory");
    asm volatile("global_load_async_to_lds_b128 %0, %1, %2"
                 :: "v"(lB0), "v"(off), "s"(baseB) : "memory");

    unsigned buf = 0;
    unsigned tn = t + gridDim.x;
    unsigned noff = off + step;
    for (;;) {
        const bool hasNext = (tn < numTiles);
        if (hasNext) {
            // Issue next tile into the other buffer, then wait for current
            // tile only (async loads complete in order): ASYNCcnt <= 2.
            const unsigned la = buf ? lA0 : lA1;
            const unsigned lb = buf ? lB0 : lB1;
            asm volatile("global_load_async_to_lds_b128 %0, %1, %2"
                         :: "v"(la), "v"(noff), "s"(baseA) : "memory");
            asm volatile("global_load_async_to_lds_b128 %0, %1, %2"
                         :: "v"(lb), "v"(noff), "s"(baseB) : "memory");
            asm volatile("s_wait_asynccnt 0x2" ::: "memory");
        } else {
            asm volatile("s_wait_asynccnt 0x0" ::: "memory");
        }

        const v4f x = sA[buf][tid];   // ds_load_b128
        const v4f y = sB[buf][tid];   // ds_load_b128
        v4f r;
        r.x = ds_comb(x.x, y.x);
        r.y = ds_comb(x.y, y.y);
        r.z = ds_comb(x.z, y.z);
        r.w = ds_comb(x.w, y.w);
        // Write-once output: nontemporal B128 store, keep L2 for input stream
        __builtin_nontemporal_store(
            r, reinterpret_cast<v4f*>(reinterpret_cast<char*>(out) + off));

        if (!hasNext) break;
        off = noff;
        noff += step;
        buf ^= 1u;
        tn += gridDim.x;
    }
}

// Scalar tail for element counts not divisible by BLOCK*4 (zero work for the
// reference shape: 2097152*21 = 44,040,192 = 43008 * 1024 exactly).
__global__ __launch_bounds__(BLOCK) void ds_tail(
    const float* __restrict__ a1, const float* __restrict__ a2,
    float* __restrict__ out, unsigned start, unsigned total)
{
    const unsigned i = start + blockIdx.x * BLOCK + threadIdx.x;
    if (i < total) out[i] = ds_comb(a1[i], a2[i]);
}

extern "C" void kernel_launch(void* const* d_in, const int* in_sizes, int n_in,
                              void* d_out, int out_size, void* d_ws, size_t ws_size,
                              hipStream_t stream) {
    (void)in_sizes; (void)n_in; (void)d_ws; (void)ws_size;
    const float* a1 = (const float*)d_in[0];
    const float* a2 = (const float*)d_in[1];
    float* out = (float*)d_out;

    const unsigned total = (unsigned)out_size;          // 44,040,192 f32
    const unsigned tileElems = BLOCK * 4u;              // 1024 f32 per tile
    const unsigned tiles = total / tileElems;           // 43,008
    const unsigned mainElems = tiles * tileElems;

    if (tiles) {
        const unsigned grid = tiles < 6144u ? tiles : 6144u;  // 7 tiles/block
        ds_main<<<grid, BLOCK, 0, stream>>>(a1, a2, out, tiles);
    }
    if (mainElems < total) {
        const unsigned rem = total - mainElems;
        ds_tail<<<(rem + BLOCK - 1) / BLOCK, BLOCK, 0, stream>>>(
            a1, a2, out, mainElems, total);
    }
}